// Shoot_37091337568586
// MI455X (gfx1250) — compile-verified
//
#include <hip/hip_runtime.h>
#include <math.h>

// Geodesic shooting (LDDMM "fast" scheme) on [B=2, D=3, 128^3] f32.
//   m0 = L v           -> 7-point periodic stencil (L is local in space)
//   per step: sv = dt * K m  (K = L^-1 via 3D DFT, done as WMMA-f32 GEMMs)
//             phi = pullback(phi, sv) + sv ;  m = pullback(m, sv)
//
// DFT along each axis = 128x128 DFT matrix times [128 x 98304] pencils,
// executed on V_WMMA_F32_16X16X4_F32 (f32 matrix pipe, matches reference
// precision). Inverse transform reuses the same twiddle table with a sign
// flip (conjugate); the 1/N^3 normalization is folded into the Green's
// function multiply.

#define NPT    128
#define VOL    (NPT*NPT*NPT)          // 2^21
#define BATCH  2
#define DIMS   3
#define CH     (BATCH*DIMS)           // 6
#define TOTAL  (CH*VOL)               // 12,582,912 floats = 48 MB
#define NSTEPS 8
#define ALPHA  1e-2f
#define GAMMA  1e-3f

static __device__ __host__ constexpr float PI_F = 3.14159265358979323846f;

typedef __attribute__((ext_vector_type(2))) float v2f;
typedef __attribute__((ext_vector_type(8))) float v8f;

// ---------------------------------------------------------------- utilities
__global__ void k_fill(float* __restrict__ p, float val, int n) {
    int i = blockIdx.x * blockDim.x + threadIdx.x;
    if (i < n) p[i] = val;
}

// Twiddle table: W[k][n] = exp(-2*pi*i*k*n/128). 128KB total, L2 resident.
__global__ void k_twiddle(float* __restrict__ wr, float* __restrict__ wi) {
    int idx = blockIdx.x * blockDim.x + threadIdx.x;
    if (idx >= NPT * NPT) return;
    int k = idx / NPT, n = idx % NPT;
    float ang = -2.0f * PI_F * (float)((k * n) & (NPT - 1)) / (float)NPT;
    wr[idx] = cosf(ang);
    wi[idx] = sinf(ang);
}

// m = L v : (GAMMA + 6*ALPHA)*v - ALPHA*(6 periodic neighbors).
__global__ __launch_bounds__(256) void k_lop(const float* __restrict__ v,
                                             float* __restrict__ m) {
    size_t idx = (size_t)blockIdx.x * 256 + threadIdx.x;
    if (idx >= (size_t)TOTAL) return;
    size_t c = idx >> 21;            // channel (VOL = 2^21)
    size_t r = idx & (VOL - 1);
    int x = (int)(r >> 14), y = (int)((r >> 7) & 127), z = (int)(r & 127);
    const float* f = v + c * (size_t)VOL;
    #define AT(xx, yy, zz) f[(((size_t)(xx)) << 14) | ((size_t)(yy) << 7) | (size_t)(zz)]
    float ctr = f[r];
    float nb = AT((x + 1) & 127, y, z) + AT((x + 127) & 127, y, z)
             + AT(x, (y + 1) & 127, z) + AT(x, (y + 127) & 127, z)
             + AT(x, y, (z + 1) & 127) + AT(x, y, (z + 127) & 127);
    #undef AT
    m[idx] = (GAMMA + 6.0f * ALPHA) * ctr - ALPHA * nb;
}

// ------------------------------------------------- batched pencil DFT (WMMA)
// One block = 16 pencils x full 128-point transform. 8 wave32 per block,
// each wave owns one 16(k) x 16(pencil) output tile, K-loop over n in
// chunks of 4 (V_WMMA_F32_16X16X4_F32). Complex GEMM via 4 real products:
//   Yr = Wr*Xr - sign*Wi*Xi ; Yi = Wr*Xi + sign*Wi*Xr   (sign=+1 fwd, -1 inv)
// Array viewed as [outer, 128, inner]; element n of pencil p=(o,j) at
// ((o*128 + n)*inner + j).
__global__ __launch_bounds__(256)
void k_dft(const float* __restrict__ ir, const float* __restrict__ ii,
           float* __restrict__ outr, float* __restrict__ outi,
           const float* __restrict__ wr, const float* __restrict__ wi,
           int inner, float sign, float oscale) {
    __shared__ float xr[NPT * 16];
    __shared__ float xi[NPT * 16];
    const int tid = threadIdx.x;
    const int p0 = blockIdx.x * 16;

    // Stage the 128 x 16 complex input tile into LDS (coalesced per axis).
    for (int e = 0; e < 8; ++e) {
        int t = e * 256 + tid;              // 0..2047
        int n, col;
        if (inner == 1) { n = t & (NPT - 1); col = t >> 7; }   // z axis: n fastest
        else            { col = t & 15;      n = t >> 4;    }  // x/y axis: pencil fastest
        int p = p0 + col;
        int o = p / inner, j = p - o * inner;
        size_t g = ((size_t)o * NPT + n) * (size_t)inner + (size_t)j;
        xr[n * 16 + col] = ir[g];
        xi[n * 16 + col] = ii ? ii[g] : 0.0f;
    }
    __syncthreads();

    const int wave  = tid >> 5;
    const int lane  = tid & 31;
    const int kbase = wave * 16;            // this wave's 16 output rows
    const int mrow  = lane & 15;            // A-matrix row / B-matrix column
    const int khalf = (lane >> 4) * 2;      // K sub-index 0 or 2

    v8f aRR = {}, aII = {}, aRI = {}, aIR = {};
    for (int ks = 0; ks < NPT / 4; ++ks) {
        int nb = ks * 4;
        // A fragment (16x4): W[kbase+mrow][nb+khalf+{0,1}]
        int wb = (kbase + mrow) * NPT + nb + khalf;
        v2f ar, ai, br, bi;
        ar.x = wr[wb];     ar.y = wr[wb + 1];
        ai.x = wi[wb];     ai.y = wi[wb + 1];
        // B fragment (4x16): X[nb+khalf+{0,1}][mrow] from LDS (conflict-free)
        int xb = (nb + khalf) * 16 + mrow;
        br.x = xr[xb];     br.y = xr[xb + 16];
        bi.x = xi[xb];     bi.y = xi[xb + 16];
        aRR = __builtin_amdgcn_wmma_f32_16x16x4_f32(false, ar, false, br, (short)0, aRR, false, false);
        aII = __builtin_amdgcn_wmma_f32_16x16x4_f32(false, ai, false, bi, (short)0, aII, false, false);
        aRI = __builtin_amdgcn_wmma_f32_16x16x4_f32(false, ar, false, bi, (short)0, aRI, false, false);
        aIR = __builtin_amdgcn_wmma_f32_16x16x4_f32(false, ai, false, br, (short)0, aIR, false, false);
    }

    // D tile layout: vgpr r holds M = r + 8*(lane>=16), N = lane&15.
    const int mhi  = (lane >> 4) * 8;
    const int ncol = lane & 15;
    int p = p0 + ncol;
    int o = p / inner, j = p - o * inner;
    #pragma unroll
    for (int r = 0; r < 8; ++r) {
        int kout = kbase + mhi + r;
        size_t g = ((size_t)o * NPT + kout) * (size_t)inner + (size_t)j;
        outr[g] = (aRR[r] - sign * aII[r]) * oscale;
        if (outi) outi[g] = (aRI[r] + sign * aIR[r]) * oscale;
    }
}

// Spectral Green's function: A(k) *= 1 / (lhat(k) * N^3)
__global__ __launch_bounds__(256) void k_greens(float* __restrict__ gr,
                                                float* __restrict__ gi) {
    size_t idx = (size_t)blockIdx.x * 256 + threadIdx.x;
    if (idx >= (size_t)TOTAL) return;
    size_t r = idx & (VOL - 1);
    int kx = (int)(r >> 14), ky = (int)((r >> 7) & 127), kz = (int)(r & 127);
    const float w = 2.0f * PI_F / (float)NPT;
    float lap = 6.0f - 2.0f * (cosf(w * (float)kx) + cosf(w * (float)ky) + cosf(w * (float)kz));
    float s = 1.0f / ((GAMMA + ALPHA * lap) * (float)VOL);
    gr[idx] *= s;
    gi[idx] *= s;
}

// One Euler step: phi_n = pullback(phi, sv) + sv ; m_n = pullback(m, sv).
// Periodic trilinear interpolation; coords shared by the 3 channels.
__global__ __launch_bounds__(256)
void k_step(const float* __restrict__ phi, const float* __restrict__ m,
            const float* __restrict__ sv,
            float* __restrict__ phin, float* __restrict__ mn) {
    size_t idx = (size_t)blockIdx.x * 256 + threadIdx.x;
    if (idx >= (size_t)BATCH * VOL) return;
    size_t b = idx >> 21;
    size_t r = idx & (VOL - 1);
    int x = (int)(r >> 14), y = (int)((r >> 7) & 127), z = (int)(r & 127);
    const float* svb = sv + b * (size_t)DIMS * VOL;
    float s0 = svb[0 * (size_t)VOL + r];
    float s1 = svb[1 * (size_t)VOL + r];
    float s2 = svb[2 * (size_t)VOL + r];
    float cx = (float)x + s0, cy = (float)y + s1, cz = (float)z + s2;
    float fx = floorf(cx), fy = floorf(cy), fz = floorf(cz);
    float tx = cx - fx, ty = cy - fy, tz = cz - fz;
    int x0 = ((int)fx) & 127, x1 = (x0 + 1) & 127;
    int y0 = ((int)fy) & 127, y1 = (y0 + 1) & 127;
    int z0 = ((int)fz) & 127, z1 = (z0 + 1) & 127;
    size_t o000 = (((size_t)x0 << 7) | y0) << 7 | z0;
    size_t o001 = (((size_t)x0 << 7) | y0) << 7 | z1;
    size_t o010 = (((size_t)x0 << 7) | y1) << 7 | z0;
    size_t o011 = (((size_t)x0 << 7) | y1) << 7 | z1;
    size_t o100 = (((size_t)x1 << 7) | y0) << 7 | z0;
    size_t o101 = (((size_t)x1 << 7) | y0) << 7 | z1;
    size_t o110 = (((size_t)x1 << 7) | y1) << 7 | z0;
    size_t o111 = (((size_t)x1 << 7) | y1) << 7 | z1;
    float w000 = (1.f - tx) * (1.f - ty) * (1.f - tz);
    float w001 = (1.f - tx) * (1.f - ty) * tz;
    float w010 = (1.f - tx) * ty * (1.f - tz);
    float w011 = (1.f - tx) * ty * tz;
    float w100 = tx * (1.f - ty) * (1.f - tz);
    float w101 = tx * (1.f - ty) * tz;
    float w110 = tx * ty * (1.f - tz);
    float w111 = tx * ty * tz;
    #pragma unroll
    for (int d = 0; d < DIMS; ++d) {
        size_t cb = (b * DIMS + d) * (size_t)VOL;
        const float* fp = phi + cb;
        const float* fm = m + cb;
        float pv = w000 * fp[o000] + w001 * fp[o001] + w010 * fp[o010] + w011 * fp[o011]
                 + w100 * fp[o100] + w101 * fp[o101] + w110 * fp[o110] + w111 * fp[o111];
        float mv = w000 * fm[o000] + w001 * fm[o001] + w010 * fm[o010] + w011 * fm[o011]
                 + w100 * fm[o100] + w101 * fm[o101] + w110 * fm[o110] + w111 * fm[o111];
        float svd = (d == 0) ? s0 : (d == 1) ? s1 : s2;
        phin[cb + r] = pv + svd;
        mn[cb + r]   = mv;
    }
}

// ---------------------------------------------------------------- launcher
extern "C" void kernel_launch(void* const* d_in, const int* in_sizes, int n_in,
                              void* d_out, int out_size, void* d_ws, size_t ws_size,
                              hipStream_t stream) {
    const float* v = (const float*)d_in[0];
    float* out = (float*)d_out;                 // phi ping (final result here)
    float* ws = (float*)d_ws;

    // Workspace layout (7 x 48MB field slots + 128KB twiddles ~= 336 MB):
    const size_t slot = (size_t)TOTAL;
    float* m_a   = ws + 0 * slot;
    float* m_b   = ws + 1 * slot;
    float* phi_b = ws + 2 * slot;
    float* fr    = ws + 3 * slot;   // also holds sv (real output of inverse z)
    float* fi    = ws + 4 * slot;
    float* gr    = ws + 5 * slot;
    float* gi    = ws + 6 * slot;
    float* wr    = ws + 7 * slot;
    float* wi    = wr + NPT * NPT;

    const int nPencilBlocks = (CH * NPT * NPT) / 16;   // 6144 full tiles
    const int nElemBlocks   = (TOTAL + 255) / 256;
    const float dt = 1.0f / (float)NSTEPS;

    k_twiddle<<<(NPT * NPT + 255) / 256, 256, 0, stream>>>(wr, wi);
    k_lop<<<nElemBlocks, 256, 0, stream>>>(v, m_a);
    k_fill<<<nElemBlocks, 256, 0, stream>>>(out, 0.0f, TOTAL);

    float* phiC = out;  float* phiN = phi_b;
    float* mC   = m_a;  float* mN   = m_b;

    for (int s = 0; s < NSTEPS; ++s) {
        // sv = dt * K(m): forward 3D DFT, Green's multiply, inverse 3D DFT.
        k_dft<<<nPencilBlocks, 256, 0, stream>>>(mC, nullptr, gr, gi, wr, wi, 1,        +1.0f, 1.0f);
        k_dft<<<nPencilBlocks, 256, 0, stream>>>(gr, gi, fr, fi, wr, wi, NPT,           +1.0f, 1.0f);
        k_dft<<<nPencilBlocks, 256, 0, stream>>>(fr, fi, gr, gi, wr, wi, NPT * NPT,     +1.0f, 1.0f);
        k_greens<<<nElemBlocks, 256, 0, stream>>>(gr, gi);
        k_dft<<<nPencilBlocks, 256, 0, stream>>>(gr, gi, fr, fi, wr, wi, NPT * NPT,     -1.0f, 1.0f);
        k_dft<<<nPencilBlocks, 256, 0, stream>>>(fr, fi, gr, gi, wr, wi, NPT,           -1.0f, 1.0f);
        k_dft<<<nPencilBlocks, 256, 0, stream>>>(gr, gi, fr, nullptr, wr, wi, 1,        -1.0f, dt);
        // Compose phi and transport m through id + sv.
        k_step<<<(BATCH * VOL + 255) / 256, 256, 0, stream>>>(phiC, mC, fr, phiN, mN);
        float* t;
        t = phiC; phiC = phiN; phiN = t;
        t = mC;   mC = mN;     mN = t;
    }
    // NSTEPS is even -> phiC == out after the final swap.
}